// GeometricRotorLearner_18983755448640
// MI455X (gfx1250) — compile-verified
//
#include <hip/hip_runtime.h>
#include <math.h>

#define STATE_DIM 256
#define BATCH     131072
#define NELEM     (BATCH * STATE_DIM)   /* 33,554,432 floats */
#define NVEC4     (NELEM / 4)           /* 8,388,608 float4 */
#define BLOCK     256
#define NBLOCKS   (NVEC4 / BLOCK)       /* 32,768 blocks */

typedef float v4f __attribute__((ext_vector_type(4)));
typedef int   v4i __attribute__((ext_vector_type(4)));

#define AS_GLOBAL __attribute__((address_space(1)))
#define AS_LDS    __attribute__((address_space(3)))

// ---------------------------------------------------------------------------
// Kernel 0: compute rotor scalars once (double precision internally).
// params[0] = cos(a)-1, params[1] = sin(a), params[2..7] = v_hat[0..5]
// where a = ||theta|| / 2.
// ---------------------------------------------------------------------------
__global__ void rotor_setup(const float* __restrict__ theta,
                            float* __restrict__ params) {
  if (threadIdx.x == 0 && blockIdx.x == 0) {
    double t[6], n2 = 0.0;
    for (int k = 0; k < 6; ++k) { t[k] = (double)theta[k]; n2 += t[k] * t[k]; }
    double n = sqrt(n2);
    double c1 = 0.0, st = 0.0, inv = 0.0;
    if (n > 1e-30) {
      double a = 0.5 * n;
      c1 = cos(a) - 1.0;
      st = sin(a);
      inv = 1.0 / n;
    }
    params[0] = (float)c1;
    params[1] = (float)st;
    for (int k = 0; k < 6; ++k) params[2 + k] = (float)(t[k] * inv);
  }
}

// ---------------------------------------------------------------------------
// Kernel 1: streaming rotor application + per-block MSE partial sums.
// One float4 per thread; rows are 64 float4 wide. Only c4 in {0,1} need the
// rank-2 correction (columns 0..6), which requires the row's first 8 floats.
// ---------------------------------------------------------------------------
__global__ __launch_bounds__(BLOCK) void rotor_apply(
    const float* __restrict__ A, const float* __restrict__ Bst,
    const float* __restrict__ params, float* __restrict__ Out,
    float* __restrict__ partials) {
  const int tid = threadIdx.x;
  const int f   = blockIdx.x * BLOCK + tid;       // float4 index, < 2^23

  // Broadcast rotor parameters (L2/L0-resident after first touch).
  const v4f p0 = *(const v4f*)(params);           // {c1, st, v0, v1}
  const v4f p1 = *(const v4f*)(params + 4);       // {v2, v3, v4, v5}
  const float c1 = p0.x, st = p0.y;

  const v4f* A4 = (const v4f*)A;
  const v4f* B4 = (const v4f*)Bst;
  v4f*       O4 = (v4f*)Out;

  v4f a = __builtin_nontemporal_load(A4 + f);     // th:NT — touch-once stream
  const int c4   = f & 63;                        // float4 column within row
  const int row0 = f - c4;                        // float4 index of row start

  v4f o = a;
  if (c4 < 2) {
    v4f h0, h1;                                   // h0 = x[0..3], h1 = x[4..7]
    if (c4 == 0) { h0 = a; h1 = __builtin_nontemporal_load(A4 + row0 + 1); }
    else         { h0 = __builtin_nontemporal_load(A4 + row0); h1 = a; }
    const float s  = p0.z * h0.y + p0.w * h0.z + p1.x * h0.w +
                     p1.y * h1.x + p1.z * h1.y + p1.w * h1.z;   // v_hat . x
    const float x0 = h0.x;
    const float cv = c1 * s - st * x0;            // coefficient on v_hat
    if (c4 == 0) {
      o.x += c1 * x0 + st * s;                    // column 0
      o.y += cv * p0.z;                           // columns 1..3
      o.z += cv * p0.w;
      o.w += cv * p1.x;
    } else {
      o.x += cv * p1.y;                           // columns 4..6
      o.y += cv * p1.z;
      o.z += cv * p1.w;
    }
  }
  __builtin_nontemporal_store(o, O4 + f);

  // --- state_b stream: stage through LDS via the gfx1250 async data mover
  //     (global_load_async_to_lds_b128 + s_wait_asynccnt). LDS bw >> HBM bw,
  //     so the extra hop is free in this HBM-bound kernel. ---
  v4f b;
#if __has_builtin(__builtin_amdgcn_global_load_async_to_lds_b128) && \
    __has_builtin(__builtin_amdgcn_s_wait_asynccnt)
  __shared__ v4f bstage[BLOCK];
  {
    v4i* gsrc_gen = (v4i*)(B4 + f);               // drop const, retype (generic)
    v4i* lds_gen  = (v4i*)&bstage[tid];
    __builtin_amdgcn_global_load_async_to_lds_b128(
        (AS_GLOBAL v4i*)gsrc_gen,
        (AS_LDS    v4i*)lds_gen,
        0, 0);
  }
  __builtin_amdgcn_s_wait_asynccnt(0);
  b = bstage[tid];
#else
  b = __builtin_nontemporal_load(B4 + f);
#endif

  const v4f d = o - b;
  float local = d.x * d.x + d.y * d.y + d.z * d.z + d.w * d.w;

  // Deterministic reduction: wave32 shuffle -> LDS -> per-block partial.
  #pragma unroll
  for (int off = 16; off > 0; off >>= 1)
    local += __shfl_down(local, off, 32);

  __shared__ float wsum[BLOCK / 32];
  const int wave = tid >> 5, lane = tid & 31;
  if (lane == 0) wsum[wave] = local;
  __syncthreads();
  if (wave == 0) {
    float v = (lane < (BLOCK / 32)) ? wsum[lane] : 0.0f;
    #pragma unroll
    for (int off = 4; off > 0; off >>= 1)
      v += __shfl_down(v, off, 32);
    if (lane == 0) partials[blockIdx.x] = v;
  }
}

// ---------------------------------------------------------------------------
// Kernel 2: deterministic single-block reduction of the 32768 partials.
// ---------------------------------------------------------------------------
__global__ __launch_bounds__(1024) void reduce_loss(
    const float* __restrict__ partials, float* __restrict__ loss_out) {
  __shared__ float sm[1024];
  const int tid = threadIdx.x;
  float s = 0.0f;
  for (int i = tid; i < NBLOCKS; i += 1024) s += partials[i];  // fixed order
  sm[tid] = s;
  __syncthreads();
  for (int k = 512; k > 0; k >>= 1) {
    if (tid < k) sm[tid] += sm[tid + k];
    __syncthreads();
  }
  if (tid == 0) loss_out[0] = sm[0] * (1.0f / (float)NELEM);
}

// ---------------------------------------------------------------------------
// Entry point. Inputs: state_a [131072*256], state_b [131072*256], theta [6].
// Output: transformed [33,554,432] floats followed by loss [1] float.
// Workspace: params (8 floats) at offset 0; block partials at float offset 256.
// ---------------------------------------------------------------------------
extern "C" void kernel_launch(void* const* d_in, const int* in_sizes, int n_in,
                              void* d_out, int out_size, void* d_ws,
                              size_t ws_size, hipStream_t stream) {
  (void)in_sizes; (void)n_in; (void)out_size; (void)ws_size;
  const float* A     = (const float*)d_in[0];
  const float* Bst   = (const float*)d_in[1];
  const float* theta = (const float*)d_in[2];
  float* out      = (float*)d_out;
  float* params   = (float*)d_ws;
  float* partials = (float*)d_ws + 256;  // 1 KB in: keeps params cacheline clean

  rotor_setup<<<1, 1, 0, stream>>>(theta, params);
  rotor_apply<<<NBLOCKS, BLOCK, 0, stream>>>(A, Bst, params, out, partials);
  reduce_loss<<<1, 1024, 0, stream>>>(partials, out + NELEM);
}